// GATLay_63153199120410
// MI455X (gfx1250) — compile-verified
//
#include <hip/hip_runtime.h>
#include <math.h>

typedef __attribute__((ext_vector_type(2))) float v2f;
typedef __attribute__((ext_vector_type(8))) float v8f;

#define N_NODES 50000
#define E_EDGES 1600000
#define INF_    256
#define HID     64
#define HEADS   8
#define OUTF    256
#define C1      (HEADS * HID)   // 512

__device__ __forceinline__ float fatomic_add(float* p, float v) {
    return unsafeAtomicAdd(p, v);   // native global_atomic_add_f32
}

// ---------------------------------------------------------------------------
// WMMA fp32 GEMM:  C[M x NC] = A[M x K] @ B[K x NC] + bias[NC]
// MODE 0: B element (k,c) = W_heads[(c>>6)][k][(c&63)]   (HEADS, K, 64) layout
// MODE 1: B element (k,c) = B[k*NC + c]                   row-major
// One wave per 16x16 output tile. M must be a multiple of 16; grid sized so
// every wave maps to a valid tile (no divergence -> EXEC all ones for WMMA).
// ---------------------------------------------------------------------------
template <int MODE>
__global__ __launch_bounds__(256)
void gemm_wmma_f32(const float* __restrict__ A, const float* __restrict__ B,
                   const float* __restrict__ bias, float* __restrict__ C,
                   int K, int NC) {
    int wave = blockIdx.x * (blockDim.x >> 5) + (threadIdx.x >> 5);
    int lane = threadIdx.x & 31;
    int half = lane >> 4;
    int l    = lane & 15;

    int ntilesN = NC >> 4;
    int tm = wave / ntilesN;
    int tn = wave % ntilesN;

    int row = tm * 16 + l;     // A row for this lane
    int col = tn * 16 + l;     // B/C column for this lane

    const float* arow = A + (size_t)row * K;

    v8f acc = {};
    for (int k0 = 0; k0 < K; k0 += 4) {
        int ka = k0 + 2 * half;
        v2f a, b;
        a.x = arow[ka];
        a.y = arow[ka + 1];
        if (MODE == 0) {
            int hd = col >> 6, hc = col & 63;
            const float* bp = B + ((size_t)hd * K + ka) * 64 + hc;
            b.x = bp[0];
            b.y = bp[64];
        } else {
            const float* bp = B + (size_t)ka * NC + col;
            b.x = bp[0];
            b.y = bp[NC];
        }
        acc = __builtin_amdgcn_wmma_f32_16x16x4_f32(
            /*neg_a=*/false, a, /*neg_b=*/false, b,
            /*c_mod=*/(short)0, acc, /*reuse_a=*/false, /*reuse_b=*/false);
    }

    float bc = bias[col];
    float* crow = C + (size_t)(tm * 16 + 8 * half) * NC + col;
#pragma unroll
    for (int r = 0; r < 8; ++r)
        crow[(size_t)r * NC] = acc[r] + bc;
}

// ---------------------------------------------------------------------------
// Per-node attention scalars, layer 1: one thread per (node, head)
// ---------------------------------------------------------------------------
__global__ void alpha1_kernel(const float* __restrict__ H, const float* __restrict__ a_heads,
                              float* __restrict__ asrc, float* __restrict__ adst) {
    int t = blockIdx.x * blockDim.x + threadIdx.x;
    if (t >= N_NODES * HEADS) return;
    int n = t >> 3, hd = t & 7;
    const float* hp = H + (size_t)n * C1 + hd * HID;
    const float* ap = a_heads + (size_t)hd * 2 * HID;   // (HEADS, 128, 1)
    float s = 0.f, d = 0.f;
#pragma unroll 4
    for (int i = 0; i < HID; ++i) {
        float v = hp[i];
        s += v * ap[i];
        d += v * ap[HID + i];
    }
    asrc[t] = s;
    adst[t] = d;
}

// Per-node attention scalars, layer 2: one thread per node
__global__ void alpha2_kernel(const float* __restrict__ H2, const float* __restrict__ a_out,
                              float* __restrict__ asrc, float* __restrict__ adst) {
    int n = blockIdx.x * blockDim.x + threadIdx.x;
    if (n >= N_NODES) return;
    const float* hp = H2 + (size_t)n * OUTF;
    float s = 0.f, d = 0.f;
#pragma unroll 4
    for (int i = 0; i < OUTF; ++i) {
        float v = hp[i];
        s += v * a_out[i];
        d += v * a_out[OUTF + i];
    }
    asrc[n] = s;
    adst[n] = d;
}

// ---------------------------------------------------------------------------
// Edge pass A, layer 1: w = exp(leaky_relu(asrc[row]+adst[col])); denom[row]+=w
// one thread per (edge, head)
// ---------------------------------------------------------------------------
__global__ void edge1A_kernel(const int* __restrict__ ei,
                              const float* __restrict__ asrc, const float* __restrict__ adst,
                              float* __restrict__ ex, float* __restrict__ den) {
    int t = blockIdx.x * blockDim.x + threadIdx.x;
    if (t >= E_EDGES * HEADS) return;
    int e = t >> 3, hd = t & 7;
    int r = ei[e];
    int c = ei[E_EDGES + e];
    float z = asrc[r * 8 + hd] + adst[c * 8 + hd];
    z = z > 0.f ? z : 0.01f * z;
    float w = expf(z);
    ex[t] = w;
    fatomic_add(&den[r * 8 + hd], w);
}

// Edge pass B, layer 1: Hc[col] += (ex/den[row]) * H[row]
// one thread per (edge, 8-float feature chunk): 64 chunks of C1=512
__global__ void edge1B_kernel(const int* __restrict__ ei, const float* __restrict__ H,
                              const float* __restrict__ ex, const float* __restrict__ den,
                              float* __restrict__ Hc) {
    int t = blockIdx.x * blockDim.x + threadIdx.x;   // < E*64
    int e = t >> 6, g = t & 63;
    int r = ei[e];
    int c = ei[E_EDGES + e];
    int hd = g >> 3;                                  // head of this chunk
    float w = ex[e * 8 + hd] / den[r * 8 + hd];
    const float4* hp = (const float4*)(H + (size_t)r * C1 + g * 8);
    float4 v0 = hp[0], v1 = hp[1];
    float* op = Hc + (size_t)c * C1 + g * 8;
    fatomic_add(op + 0, w * v0.x);
    fatomic_add(op + 1, w * v0.y);
    fatomic_add(op + 2, w * v0.z);
    fatomic_add(op + 3, w * v0.w);
    fatomic_add(op + 4, w * v1.x);
    fatomic_add(op + 5, w * v1.y);
    fatomic_add(op + 6, w * v1.z);
    fatomic_add(op + 7, w * v1.w);
}

// Edge pass A, layer 2: one thread per edge
__global__ void edge2A_kernel(const int* __restrict__ ei,
                              const float* __restrict__ asrc, const float* __restrict__ adst,
                              float* __restrict__ ex, float* __restrict__ den) {
    int e = blockIdx.x * blockDim.x + threadIdx.x;
    if (e >= E_EDGES) return;
    int r = ei[e];
    int c = ei[E_EDGES + e];
    float z = asrc[r] + adst[c];
    z = z > 0.f ? z : 0.01f * z;
    float w = expf(z);
    ex[e] = w;
    fatomic_add(&den[r], w);
}

// Edge pass B, layer 2: one thread per (edge, 8-float chunk): 32 chunks of 256
__global__ void edge2B_kernel(const int* __restrict__ ei, const float* __restrict__ H2,
                              const float* __restrict__ ex, const float* __restrict__ den,
                              float* __restrict__ O2) {
    int t = blockIdx.x * blockDim.x + threadIdx.x;   // < E*32
    int e = t >> 5, g = t & 31;
    int r = ei[e];
    int c = ei[E_EDGES + e];
    float w = ex[e] / den[r];
    const float4* hp = (const float4*)(H2 + (size_t)r * OUTF + g * 8);
    float4 v0 = hp[0], v1 = hp[1];
    float* op = O2 + (size_t)c * OUTF + g * 8;
    fatomic_add(op + 0, w * v0.x);
    fatomic_add(op + 1, w * v0.y);
    fatomic_add(op + 2, w * v0.z);
    fatomic_add(op + 3, w * v0.w);
    fatomic_add(op + 4, w * v1.x);
    fatomic_add(op + 5, w * v1.y);
    fatomic_add(op + 6, w * v1.z);
    fatomic_add(op + 7, w * v1.w);
}

// ELU + row-softmax: one block (256 threads) per node, one feature per thread
__global__ __launch_bounds__(256)
void finalize_kernel(const float* __restrict__ O2, float* __restrict__ out) {
    __shared__ float red[256];
    int n = blockIdx.x;
    int t = threadIdx.x;
    float v = O2[(size_t)n * OUTF + t];
    v = v > 0.f ? v : (expf(v) - 1.0f);          // ELU
    red[t] = v;
    __syncthreads();
    for (int s = 128; s > 0; s >>= 1) {
        if (t < s) red[t] = fmaxf(red[t], red[t + s]);
        __syncthreads();
    }
    float mx = red[0];
    __syncthreads();
    float ev = expf(v - mx);
    red[t] = ev;
    __syncthreads();
    for (int s = 128; s > 0; s >>= 1) {
        if (t < s) red[t] += red[t + s];
        __syncthreads();
    }
    out[(size_t)n * OUTF + t] = ev / red[0];
}

// ---------------------------------------------------------------------------
extern "C" void kernel_launch(void* const* d_in, const int* in_sizes, int n_in,
                              void* d_out, int out_size, void* d_ws, size_t ws_size,
                              hipStream_t stream) {
    const float* x  = (const float*)d_in[0];
    const int*   ei = (const int*)d_in[1];
    // d_in[2] = dropout (unused, == 0)
    const float* W1 = (const float*)d_in[3];   // (HEADS, 256, 64)
    const float* b1 = (const float*)d_in[4];   // (HEADS, 64) -> flat 512, matches c index
    const float* a1 = (const float*)d_in[5];   // (HEADS, 128, 1)
    const float* W2 = (const float*)d_in[6];   // (512, 256)
    const float* b2 = (const float*)d_in[7];   // (256,)
    const float* a2 = (const float*)d_in[8];   // (512, 1)
    float* out = (float*)d_out;
    float* ws  = (float*)d_ws;

    // workspace layout (floats)
    size_t oA    = 0;                               // N*512 : H1, later H2 (first half) + O2 (second half)
    size_t oB    = oA + (size_t)N_NODES * C1;       // N*512 : Hc accumulator
    size_t oC    = oB + (size_t)N_NODES * C1;       // E*8   : ex1, later ex2 (first E)
    size_t oS1   = oC + (size_t)E_EDGES * HEADS;    // N*8
    size_t oD1   = oS1 + (size_t)N_NODES * HEADS;   // N*8
    size_t oDen1 = oD1 + (size_t)N_NODES * HEADS;   // N*8
    size_t oS2   = oDen1 + (size_t)N_NODES * HEADS; // N
    size_t oD2   = oS2 + N_NODES;                   // N
    size_t oDen2 = oD2 + N_NODES;                   // N

    float* H1    = ws + oA;
    float* Hc    = ws + oB;
    float* EX    = ws + oC;
    float* asrc1 = ws + oS1;
    float* adst1 = ws + oD1;
    float* den1  = ws + oDen1;
    float* asrc2 = ws + oS2;
    float* adst2 = ws + oD2;
    float* den2  = ws + oDen2;
    float* H2    = ws + oA;                          // reuses H1 region (dead by then)
    float* O2    = ws + oA + (size_t)N_NODES * OUTF; // second half of region A

    // ---- layer 1 ----
    hipMemsetAsync(Hc,   0, (size_t)N_NODES * C1 * sizeof(float), stream);
    hipMemsetAsync(den1, 0, (size_t)N_NODES * HEADS * sizeof(float), stream);

    // H1 = x @ W_all + b : tiles = (50000/16)*(512/16) = 3125*32 = 100000 waves
    gemm_wmma_f32<0><<<100000 / 8, 256, 0, stream>>>(x, W1, b1, H1, INF_, C1);

    alpha1_kernel<<<(N_NODES * HEADS + 255) / 256, 256, 0, stream>>>(H1, a1, asrc1, adst1);
    edge1A_kernel<<<(E_EDGES * HEADS + 255) / 256, 256, 0, stream>>>(ei, asrc1, adst1, EX, den1);
    edge1B_kernel<<<(E_EDGES * 64) / 256, 256, 0, stream>>>(ei, H1, EX, den1, Hc);

    // ---- layer 2 ----
    // H2 = Hc @ W_out + b_out : tiles = 3125*16 = 50000 waves
    gemm_wmma_f32<1><<<50000 / 8, 256, 0, stream>>>(Hc, W2, b2, H2, C1, OUTF);

    hipMemsetAsync(O2,   0, (size_t)N_NODES * OUTF * sizeof(float), stream);
    hipMemsetAsync(den2, 0, (size_t)N_NODES * sizeof(float), stream);

    alpha2_kernel<<<(N_NODES + 255) / 256, 256, 0, stream>>>(H2, a2, asrc2, adst2);
    edge2A_kernel<<<(E_EDGES + 255) / 256, 256, 0, stream>>>(ei, asrc2, adst2, EX, den2);
    edge2B_kernel<<<(E_EDGES * 32) / 256, 256, 0, stream>>>(ei, H2, EX, den2, O2);

    finalize_kernel<<<N_NODES, 256, 0, stream>>>(O2, out);
}